// VIDLossWithMask_1898375545710
// MI455X (gfx1250) — compile-verified
//
#include <hip/hip_runtime.h>
#include <math.h>

// ---- problem constants (from reference) ----
#define BB   8
#define CIN  256
#define CMID 128
#define CT   256
#define NCLS 19
#define HH   128
#define WW   128
#define HWIM (HH * WW)          // 16384 pixels per image per channel
#define TILE_N 64               // pixels per workgroup
#define TILES_PER_IMG (HWIM / TILE_N)   // 256
#define NBLOCKS (BB * TILES_PER_IMG)    // 2048
#define LDSTR 72                // padded LDS channel stride (floats); 72*4B is 16B-aligned

typedef float v2f __attribute__((ext_vector_type(2)));
typedef float v8f __attribute__((ext_vector_type(8)));

// D(16x16,f32) = A(16x4,f32) x B(4x16,f32) + C
__device__ __forceinline__ v8f wmma4(v2f a, v2f b, v8f c) {
    return __builtin_amdgcn_wmma_f32_16x16x4_f32(
        /*neg_a=*/false, a, /*neg_b=*/false, b,
        /*c_mod=*/(short)0, c, /*reuse_a=*/false, /*reuse_b=*/false);
}

// generic (__shared__) pointer -> wave-relative LDS byte offset
__device__ __forceinline__ unsigned lds_off(const void* p) {
    return (unsigned)(uintptr_t)p;
}

// async direct global -> LDS copy of 16 bytes per lane (ASYNCcnt-tracked)
__device__ __forceinline__ void async_g2l_b128(unsigned ldsaddr, const float* gaddr) {
    asm volatile("global_load_async_to_lds_b128 %0, %1, off"
                 :: "v"(ldsaddr), "v"(gaddr) : "memory");
}

__device__ __forceinline__ void wait_asynccnt0() {
    asm volatile("s_wait_asynccnt 0" ::: "memory");
}

// LDS partition (floats)
#define OFF_X   0                        // 256 x 72  : X tile, later reused as A3 (256 x 72)
#define OFF_A1  (OFF_X  + CIN  * LDSTR)  // 128 x 72
#define OFF_A2  (OFF_A1 + CMID * LDSTR)  // 128 x 72
#define OFF_T   (OFF_A2 + CMID * LDSTR)  // 256 x 72  : target tile
#define OFF_RED (OFF_T  + CT   * LDSTR)  // 8 floats
#define SMEM_FLOATS (OFF_RED + 8)
#define SMEM_BYTES  (SMEM_FLOATS * sizeof(float))

__global__ __launch_bounds__(256)
void vid_loss_main(const float* __restrict__ inp,     // [B,256,H,W]
                   const float* __restrict__ tgt,     // [B,256,H,W]
                   const int*   __restrict__ labels,  // [B,512,512]
                   const float* __restrict__ w1,      // [128,256]
                   const float* __restrict__ w2,      // [128,128]
                   const float* __restrict__ w3,      // [256,128]
                   const float* __restrict__ w4,      // [19,256]
                   const float* __restrict__ tw,      // [19,256]
                   const float* __restrict__ log_scale, // [19]
                   float* __restrict__ part)          // [NBLOCKS] partial d2 sums
{
    extern __shared__ float smem[];
    float* sX  = smem + OFF_X;   // and sA3 later
    float* sA1 = smem + OFF_A1;
    float* sA2 = smem + OFF_A2;
    float* sT  = smem + OFF_T;
    float* sRed= smem + OFF_RED;

    const int tid  = threadIdx.x;
    const int lane = tid & 31;
    const int wv   = tid >> 5;      // wave id 0..7
    const int l    = lane & 15;     // lane within half
    const int half = lane >> 4;     // 0 / 1

    const int b   = blockIdx.x / TILES_PER_IMG;
    const int hw0 = (blockIdx.x % TILES_PER_IMG) * TILE_N;

    // ---------------- phase 0: async-stage X and target tiles into LDS ----------------
    // 256 channels x 64 pixels, contiguous per channel -> one B128 async copy per lane-iter
    for (int i = tid; i < CIN * (TILE_N / 4); i += 256) {
        const int c = i >> 4;              // channel
        const int q = (i & 15) << 2;       // pixel offset in tile (multiple of 4)
        const size_t g = ((size_t)(b * CIN + c) << 14) + (size_t)hw0 + (size_t)q;
        async_g2l_b128(lds_off(&sX[c * LDSTR + q]), inp + g);
        async_g2l_b128(lds_off(&sT[c * LDSTR + q]), tgt + g);
    }
    wait_asynccnt0();
    __syncthreads();

    // 2(M) x 2(N) subtiles per wave for GEMM1/2, 4(M) x 2(N) for GEMM3
    const int ngrp = wv & 1;        // covers N-subtiles {2*ngrp, 2*ngrp+1}
    const int mgrp = wv >> 1;       // covers a group of M-subtiles

    // ---------------- GEMM1: A1 = relu(W1[128,256] * X) ----------------
    {
        v8f acc[2][2] = {};
        for (int k = 0; k < CIN; k += 4) {
            v2f bf[2], af[2];
            #pragma unroll
            for (int nn = 0; nn < 2; ++nn) {
                const int n0 = (ngrp * 2 + nn) * 16;
                bf[nn].x = sX[(k +     2 * half) * LDSTR + n0 + l];
                bf[nn].y = sX[(k + 1 + 2 * half) * LDSTR + n0 + l];
            }
            #pragma unroll
            for (int mm = 0; mm < 2; ++mm) {
                const int m = (mgrp * 2 + mm) * 16 + l;
                const float2 wvv = *reinterpret_cast<const float2*>(w1 + (size_t)m * CIN + k + 2 * half);
                af[mm].x = wvv.x; af[mm].y = wvv.y;
            }
            #pragma unroll
            for (int mm = 0; mm < 2; ++mm)
                #pragma unroll
                for (int nn = 0; nn < 2; ++nn)
                    acc[mm][nn] = wmma4(af[mm], bf[nn], acc[mm][nn]);
        }
        #pragma unroll
        for (int mm = 0; mm < 2; ++mm)
            #pragma unroll
            for (int nn = 0; nn < 2; ++nn)
                #pragma unroll
                for (int r = 0; r < 8; ++r) {
                    const float v = acc[mm][nn][r];
                    sA1[((mgrp * 2 + mm) * 16 + r + 8 * half) * LDSTR + (ngrp * 2 + nn) * 16 + l] =
                        v > 0.f ? v : 0.f;
                }
    }
    __syncthreads();

    // ---------------- GEMM2: A2 = relu(W2[128,128] * A1) ----------------
    {
        v8f acc[2][2] = {};
        for (int k = 0; k < CMID; k += 4) {
            v2f bf[2], af[2];
            #pragma unroll
            for (int nn = 0; nn < 2; ++nn) {
                const int n0 = (ngrp * 2 + nn) * 16;
                bf[nn].x = sA1[(k +     2 * half) * LDSTR + n0 + l];
                bf[nn].y = sA1[(k + 1 + 2 * half) * LDSTR + n0 + l];
            }
            #pragma unroll
            for (int mm = 0; mm < 2; ++mm) {
                const int m = (mgrp * 2 + mm) * 16 + l;
                const float2 wvv = *reinterpret_cast<const float2*>(w2 + (size_t)m * CMID + k + 2 * half);
                af[mm].x = wvv.x; af[mm].y = wvv.y;
            }
            #pragma unroll
            for (int mm = 0; mm < 2; ++mm)
                #pragma unroll
                for (int nn = 0; nn < 2; ++nn)
                    acc[mm][nn] = wmma4(af[mm], bf[nn], acc[mm][nn]);
        }
        #pragma unroll
        for (int mm = 0; mm < 2; ++mm)
            #pragma unroll
            for (int nn = 0; nn < 2; ++nn)
                #pragma unroll
                for (int r = 0; r < 8; ++r) {
                    const float v = acc[mm][nn][r];
                    sA2[((mgrp * 2 + mm) * 16 + r + 8 * half) * LDSTR + (ngrp * 2 + nn) * 16 + l] =
                        v > 0.f ? v : 0.f;
                }
    }
    __syncthreads();

    // ---------------- GEMM3: A3 = relu(W3[256,128] * A2), A3 reuses sX ----------------
    {
        v8f acc[4][2] = {};
        for (int k = 0; k < CMID; k += 4) {
            v2f bf[2], af[4];
            #pragma unroll
            for (int nn = 0; nn < 2; ++nn) {
                const int n0 = (ngrp * 2 + nn) * 16;
                bf[nn].x = sA2[(k +     2 * half) * LDSTR + n0 + l];
                bf[nn].y = sA2[(k + 1 + 2 * half) * LDSTR + n0 + l];
            }
            #pragma unroll
            for (int mm = 0; mm < 4; ++mm) {
                const int m = (mgrp * 4 + mm) * 16 + l;
                const float2 wvv = *reinterpret_cast<const float2*>(w3 + (size_t)m * CMID + k + 2 * half);
                af[mm].x = wvv.x; af[mm].y = wvv.y;
            }
            #pragma unroll
            for (int mm = 0; mm < 4; ++mm)
                #pragma unroll
                for (int nn = 0; nn < 2; ++nn)
                    acc[mm][nn] = wmma4(af[mm], bf[nn], acc[mm][nn]);
        }
        __syncthreads();   // everyone done reading old sX/sA2 before overwriting sX
        #pragma unroll
        for (int mm = 0; mm < 4; ++mm)
            #pragma unroll
            for (int nn = 0; nn < 2; ++nn)
                #pragma unroll
                for (int r = 0; r < 8; ++r) {
                    const float v = acc[mm][nn][r];
                    sX[((mgrp * 4 + mm) * 16 + r + 8 * half) * LDSTR + (ngrp * 2 + nn) * 16 + l] =
                        v > 0.f ? v : 0.f;
                }
    }
    __syncthreads();

    // ---------------- GEMM4: pred_mean = W4[19,256]*A3 ; reduced_target = TW[19,256]*T ----------------
    {
        const int o0 = (wv >> 2) * 16;          // 0 or 16 (rows 19..31 are padding, masked later)
        const int n0 = (wv & 3) * 16;           // N-subtile for this phase
        const bool arow_ok = (o0 + l) < NCLS;   // A-fragment row validity
        v8f accP = {}, accT = {};
        for (int k = 0; k < CT; k += 4) {
            v2f bP, bT;
            bP.x = sX[(k +     2 * half) * LDSTR + n0 + l];
            bP.y = sX[(k + 1 + 2 * half) * LDSTR + n0 + l];
            bT.x = sT[(k +     2 * half) * LDSTR + n0 + l];
            bT.y = sT[(k + 1 + 2 * half) * LDSTR + n0 + l];
            v2f aP, aT;
            if (arow_ok) {
                const float2 a4 = *reinterpret_cast<const float2*>(w4 + (size_t)(o0 + l) * CT + k + 2 * half);
                const float2 at = *reinterpret_cast<const float2*>(tw + (size_t)(o0 + l) * CT + k + 2 * half);
                aP.x = a4.x; aP.y = a4.y; aT.x = at.x; aT.y = at.y;
            } else {
                aP.x = 0.f; aP.y = 0.f; aT.x = 0.f; aT.y = 0.f;
            }
            accP = wmma4(aP, bP, accP);
            accT = wmma4(aT, bT, accT);
        }

        // masked squared-error accumulation; each lane owns one pixel column n = n0+l
        const int n  = n0 + l;
        const int hw = hw0 + n;
        const int hh = hw >> 7;
        const int ww = hw & 127;
        const int lab = labels[(size_t)b * (512 * 512) + (size_t)(hh << 2) * 512 + (size_t)(ww << 2)];
        const float valid = (lab >= 0 && lab < NCLS) ? 1.f : 0.f;

        float psum = 0.f;
        #pragma unroll
        for (int r = 0; r < 8; ++r) {
            const int o = o0 + r + 8 * half;
            if (o < NCLS) {
                const float ls   = log_scale[o];
                const float var  = log1pf(expf(ls)) + 1e-5f;   // softplus + eps
                const float diff = accP[r] - accT[r];
                psum += diff * diff / var;
            }
        }
        psum *= valid;

        // wave32 reduce
        #pragma unroll
        for (int off = 16; off > 0; off >>= 1)
            psum += __shfl_xor(psum, off, 32);
        if (lane == 0) sRed[wv] = psum;
        __syncthreads();
        if (tid == 0) {
            float blocksum = 0.f;
            #pragma unroll
            for (int i = 0; i < 8; ++i) blocksum += sRed[i];
            part[blockIdx.x] = blocksum;
        }
    }
}

__global__ __launch_bounds__(256)
void vid_loss_reduce(const float* __restrict__ part,
                     const float* __restrict__ log_scale,
                     float* __restrict__ out)
{
    __shared__ float red[256];
    float s = 0.f;
    for (int i = threadIdx.x; i < NBLOCKS; i += 256) s += part[i];
    red[threadIdx.x] = s;
    __syncthreads();
    for (int step = 128; step > 0; step >>= 1) {
        if (threadIdx.x < step) red[threadIdx.x] += red[threadIdx.x + step];
        __syncthreads();
    }
    if (threadIdx.x == 0) {
        float lv = 0.f;
        for (int c = 0; c < NCLS; ++c) {
            const float var = log1pf(expf(log_scale[c])) + 1e-5f;
            lv += logf(var);
        }
        // 0.5 * mean(d2*valid)/C + 0.5 * mean(log var)
        const float scale = 0.5f / ((float)BB * (float)NCLS * (float)HWIM * (float)NCLS);
        out[0] = red[0] * scale + 0.5f * lv / (float)NCLS;
    }
}

extern "C" void kernel_launch(void* const* d_in, const int* in_sizes, int n_in,
                              void* d_out, int out_size, void* d_ws, size_t ws_size,
                              hipStream_t stream) {
    const float* inp       = (const float*)d_in[0];
    const float* tgt       = (const float*)d_in[1];
    const int*   labels    = (const int*)  d_in[2];
    const float* w1        = (const float*)d_in[3];
    const float* w2        = (const float*)d_in[4];
    const float* w3        = (const float*)d_in[5];
    const float* w4        = (const float*)d_in[6];
    const float* tw        = (const float*)d_in[7];
    const float* log_scale = (const float*)d_in[8];
    float* out  = (float*)d_out;
    float* part = (float*)d_ws;   // NBLOCKS floats of scratch

    vid_loss_main<<<NBLOCKS, 256, SMEM_BYTES, stream>>>(
        inp, tgt, labels, w1, w2, w3, w4, tw, log_scale, part);
    vid_loss_reduce<<<1, 256, 0, stream>>>(part, log_scale, out);
}